// DasModule_773094113831
// MI455X (gfx1250) — compile-verified
//
#include <hip/hip_runtime.h>

// Delay-and-sum beamforming for MI455X (gfx1250).
// x:(1,128,2048) f32, sensors:(128,2) f32, grid_pts:(262144,2) f32 -> out:(262144) f32
//
// Strategy: signal traces are L2-resident (1 MB total). Stage each sensor's
// 8 KB trace into LDS with gfx1250 async global->LDS DMA (ASYNCcnt), double
// buffered, and do the data-dependent 2-tap interpolation gathers from LDS.

#define NS        128
#define NT        2048
#define NPIX      (512 * 512)
#define BLOCK     256          // 8 wave32 waves
#define PPT       4            // pixels per thread
#define PIX_PER_BLOCK (BLOCK * PPT)          // 1024
#define NBLOCKS   (NPIX / PIX_PER_BLOCK)     // 256

#define INV_CDT   (1.0f / (1500.0f * 4e-8f)) // 1/(C*DT), T0 == 0
#define IDX_MAX   2046.0f                    // T_MAX/DT = NT-2

#if __has_builtin(__builtin_amdgcn_global_load_async_to_lds_b128) && \
    __has_builtin(__builtin_amdgcn_s_wait_asynccnt)
#define USE_ASYNC 1
#else
#define USE_ASYNC 0
#endif

#if USE_ASYNC
// Builtin prototype (from clang diagnostic):
//   void __builtin_amdgcn_global_load_async_to_lds_b128(
//       v4i addrspace(1)* src, v4i addrspace(3)* dst, imm offset, imm cpol)
typedef int v4i __attribute__((__vector_size__(16)));
typedef __attribute__((address_space(1))) v4i g_v4i;  // global
typedef __attribute__((address_space(3))) v4i l_v4i;  // LDS

__device__ __forceinline__ void async_cp16(const float* g, float* l) {
  __builtin_amdgcn_global_load_async_to_lds_b128((g_v4i*)g, (l_v4i*)l, 0, 0);
}
#endif

__global__ __launch_bounds__(BLOCK)
void das_kernel(const float* __restrict__ x,        // (NS, NT)
                const float* __restrict__ sensors,  // (NS, 2)
                const float* __restrict__ grid,     // (NPIX, 2)
                float* __restrict__ out)            // (NPIX)
{
  __shared__ float sbuf[2][NT];   // 16 KB double buffer (of 320 KB/WGP)

  const int t     = threadIdx.x;
  const int pbase = blockIdx.x * PIX_PER_BLOCK;

  float px[PPT], py[PPT], acc[PPT];
#pragma unroll
  for (int k = 0; k < PPT; ++k) {
    const int p = pbase + k * BLOCK + t;
    const float2 g = ((const float2*)grid)[p];   // coalesced global_load_b64
    px[k] = g.x; py[k] = g.y; acc[k] = 0.0f;
  }

#if USE_ASYNC
  // Prologue: kick off sensor 0's trace into buffer 0.
  // 8 KB / (256 threads * 16 B) = 2 b128 async loads per thread.
  {
    const float* g0 = x + t * 4;
    async_cp16(g0,        &sbuf[0][t * 4]);
    async_cp16(g0 + 1024, &sbuf[0][t * 4 + 1024]);
  }
#endif

  for (int s = 0; s < NS; ++s) {
    const int cur = s & 1;

#if USE_ASYNC
    if (s + 1 < NS) {
      // Prefetch next sensor's trace into the other buffer (async DMA).
      const float* g = x + (s + 1) * NT + t * 4;
      async_cp16(g,        &sbuf[cur ^ 1][t * 4]);
      async_cp16(g + 1024, &sbuf[cur ^ 1][t * 4 + 1024]);
      // ASYNCcnt loads retire in order: <=2 outstanding means the two loads
      // for sensor s (issued last iteration) have landed in LDS.
      __builtin_amdgcn_s_wait_asynccnt(2);
    } else {
      __builtin_amdgcn_s_wait_asynccnt(0);
    }
    __syncthreads();   // make every wave's staged data visible to all waves
#else
    __syncthreads();   // previous readers of sbuf[cur] are done
    {
      const float* g = x + s * NT + t * 4;
      const float4 v0 = ((const float4*)g)[0];
      const float4 v1 = ((const float4*)(g + 1024))[0];
      *(float4*)&sbuf[cur][t * 4]        = v0;
      *(float4*)&sbuf[cur][t * 4 + 1024] = v1;
    }
    __syncthreads();
#endif

    // Sensor position is wave-uniform -> scalar loads (KMcnt path).
    const float sx = sensors[2 * s];
    const float sy = sensors[2 * s + 1];
    const float* trace = sbuf[cur];

#pragma unroll
    for (int k = 0; k < PPT; ++k) {
      const float dx   = px[k] - sx;
      const float dy   = py[k] - sy;
      const float dist = sqrtf(fmaf(dx, dx, dy * dy));
      float idx = dist * INV_CDT;                 // (dist/C + T0)/DT, T0 = 0
      idx = (idx > IDX_MAX) ? 0.0f : idx;         // idx >= 0 always
      const float d0 = floorf(idx);
      const int   i0 = (int)d0;
      const float w0 = idx - d0;                  // weight on floor sample
      const float w1 = 1.0f - w0;                 // weight on ceil sample
      const float y0 = trace[i0];                 // ds_load (adjacent pair ->
      const float y1 = trace[i0 + 1];             //  ds_load_2addr candidate)
      acc[k] = fmaf(w0, y0, fmaf(w1, y1, acc[k]));
    }

#if USE_ASYNC
    __syncthreads();   // all reads of sbuf[cur] done before DMA overwrites it
#endif
  }

#pragma unroll
  for (int k = 0; k < PPT; ++k)
    out[pbase + k * BLOCK + t] = acc[k];          // coalesced stores
}

extern "C" void kernel_launch(void* const* d_in, const int* in_sizes, int n_in,
                              void* d_out, int out_size, void* d_ws, size_t ws_size,
                              hipStream_t stream) {
  const float* x       = (const float*)d_in[0];
  const float* sensors = (const float*)d_in[1];
  const float* grid    = (const float*)d_in[2];
  float* out           = (float*)d_out;
  das_kernel<<<NBLOCKS, BLOCK, 0, stream>>>(x, sensors, grid, out);
}